// InterpretableMultiHeadAttention_69174743269562
// MI455X (gfx1250) — compile-verified
//
#include <hip/hip_runtime.h>
#include <math.h>

// Problem constants (match reference)
#define B_   4
#define T_   512
#define S_   512
#define H_   4
#define DH_  64
#define DM_  256
#define SCALE_ 0.125f   // 1/sqrt(64)

typedef float v2f __attribute__((ext_vector_type(2)));
typedef float v8f __attribute__((ext_vector_type(8)));

#if __has_builtin(__builtin_amdgcn_tanhf)
__device__ __forceinline__ float fast_tanh(float x) { return __builtin_amdgcn_tanhf(x); }
#else
__device__ __forceinline__ float fast_tanh(float x) { return tanhf(x); }
#endif

// ---------------------------------------------------------------------------
// Generic f32 WMMA GEMM: C[M,N] = A[M,K] * op(B), one 16x16 tile per wave,
// 4 waves per block. BT=true: B stored (N,K) row-major -> C = A * B^T.
// Layouts per CDNA5 ISA (wave32):
//   A 16x4: lane<16 -> M=lane, K={k0,k0+1}; lane>=16 -> K={k0+2,k0+3}
//   B 4x16: lane<16 -> N=lane, rows K=k0,k0+1; lane>=16 -> rows K=k0+2,k0+3
//   C 16x16: VGPR v -> row v+8*half, col lane%16
// ---------------------------------------------------------------------------
template <bool BT>
__global__ __launch_bounds__(128) void gemm_wmma_f32(
    const float* __restrict__ A, const float* __restrict__ Bm,
    float* __restrict__ C, int M, int N, int K, int lda, int ldb, int ldc)
{
    const int wave = threadIdx.x >> 5;
    const int lane = threadIdx.x & 31;
    const int hi   = lane >> 4;       // half-wave select
    const int lm   = lane & 15;

    const int tiles_n = N >> 4;
    const int tile = blockIdx.x * 4 + wave;
    const int tm = tile / tiles_n;
    const int tn = tile % tiles_n;
    const int m0 = tm << 4, n0 = tn << 4;

    v8f acc = {};
    const float* Arow = A + (size_t)(m0 + lm) * lda;
    for (int k0 = 0; k0 < K; k0 += 4) {
        const int ka = k0 + 2 * hi;
        v2f a; a.x = Arow[ka]; a.y = Arow[ka + 1];
        v2f b;
        if (BT) {                       // B[n,k] (weights stored (out,in))
            const float* Bp = Bm + (size_t)(n0 + lm) * ldb + ka;
            b.x = Bp[0]; b.y = Bp[1];
        } else {                        // B[k,n]
            const float* Bp = Bm + (size_t)ka * ldb + (n0 + lm);
            b.x = Bp[0]; b.y = Bp[ldb];
        }
        acc = __builtin_amdgcn_wmma_f32_16x16x4_f32(
            false, a, false, b, (short)0, acc, false, false);
    }
    float* Crow = C + (size_t)m0 * ldc + n0 + lm;
    #pragma unroll
    for (int v = 0; v < 8; ++v)
        Crow[(size_t)(v + 8 * hi) * ldc] = acc[v];
}

// ---------------------------------------------------------------------------
// Additive-attention scores + softmax. One block per (b,h,t) row; 256 threads,
// each owns 2 s-columns. q-row and v_a cached in LDS; K streamed as float4.
// Writes normalized attention row directly into the attn slice of d_out.
// ---------------------------------------------------------------------------
__global__ __launch_bounds__(256) void scores_softmax(
    const float* __restrict__ Q, const float* __restrict__ Kp,
    const float* __restrict__ va, float* __restrict__ attn)
{
    __shared__ float q_s[DH_];
    __shared__ float va_s[DH_];
    __shared__ float red[256];

    const int tid = threadIdx.x;
    const int t  = blockIdx.x;
    const int bh = blockIdx.y;
    const int b = bh / H_, h = bh % H_;

    if (tid < DH_) {
        q_s[tid]  = Q[((size_t)(b * T_ + t)) * DM_ + h * DH_ + tid];
        va_s[tid] = va[h * DH_ + tid];
    }
    __syncthreads();

    const int s0 = tid, s1 = tid + 256;
    const float* K0 = Kp + ((size_t)(b * S_ + s0)) * DM_ + h * DH_;
    const float* K1 = K0 + (size_t)256 * DM_;

    float acc0 = 0.f, acc1 = 0.f;
    #pragma unroll 4
    for (int d = 0; d < DH_; d += 4) {
        float4 k0 = *(const float4*)(K0 + d);
        float4 k1 = *(const float4*)(K1 + d);
        float q0 = q_s[d], q1 = q_s[d+1], q2 = q_s[d+2], q3 = q_s[d+3];
        float w0 = va_s[d], w1 = va_s[d+1], w2 = va_s[d+2], w3 = va_s[d+3];
        acc0 += fast_tanh(q0 + k0.x) * w0 + fast_tanh(q1 + k0.y) * w1
              + fast_tanh(q2 + k0.z) * w2 + fast_tanh(q3 + k0.w) * w3;
        acc1 += fast_tanh(q0 + k1.x) * w0 + fast_tanh(q1 + k1.y) * w1
              + fast_tanh(q2 + k1.z) * w2 + fast_tanh(q3 + k1.w) * w3;
    }
    const float sc0 = acc0 * SCALE_, sc1 = acc1 * SCALE_;

    // row max
    red[tid] = fmaxf(sc0, sc1);
    __syncthreads();
    for (int off = 128; off > 0; off >>= 1) {
        if (tid < off) red[tid] = fmaxf(red[tid], red[tid + off]);
        __syncthreads();
    }
    const float rowmax = red[0];
    __syncthreads();

    const float e0 = __expf(sc0 - rowmax);
    const float e1 = __expf(sc1 - rowmax);
    red[tid] = e0 + e1;
    __syncthreads();
    for (int off = 128; off > 0; off >>= 1) {
        if (tid < off) red[tid] += red[tid + off];
        __syncthreads();
    }
    const float inv = 1.0f / red[0];

    float* row = attn + ((size_t)bh * T_ + t) * S_;
    row[s0] = e0 * inv;
    row[s1] = e1 * inv;
}

// ---------------------------------------------------------------------------
// attended[b,t,h*64+n] = sum_s attn[b,h,t,s] * V[b,s,h*64+n]
// Per (b,h): (512x512)@(512x64) GEMM with f32 WMMA. Block = 4 waves; wave w
// handles n-tile w (Dh/16 = 4). Output written head-interleaved -> (B,T,256).
// ---------------------------------------------------------------------------
__global__ __launch_bounds__(128) void attended_wmma(
    const float* __restrict__ attn, const float* __restrict__ V,
    float* __restrict__ Aout)
{
    const int wave = threadIdx.x >> 5;     // n-tile 0..3
    const int lane = threadIdx.x & 31;
    const int hi = lane >> 4, lm = lane & 15;
    const int mt = blockIdx.x;             // 0..T/16-1
    const int bh = blockIdx.y;             // 0..B*H-1
    const int b = bh / H_, h = bh % H_;
    const int m0 = mt << 4;
    const int n0 = wave << 4;

    const float* Arow  = attn + ((size_t)bh * T_ + m0 + lm) * S_;
    const float* Vbase = V + (size_t)b * S_ * DM_ + h * DH_;

    v8f acc = {};
    for (int k0 = 0; k0 < S_; k0 += 4) {
        const int ka = k0 + 2 * hi;
        v2f a; a.x = Arow[ka]; a.y = Arow[ka + 1];
        const float* Bp = Vbase + (size_t)ka * DM_ + n0 + lm;
        v2f bv; bv.x = Bp[0]; bv.y = Bp[DM_];
        acc = __builtin_amdgcn_wmma_f32_16x16x4_f32(
            false, a, false, bv, (short)0, acc, false, false);
    }
    float* Crow = Aout + ((size_t)b * T_ + m0) * DM_ + h * DH_ + n0 + lm;
    #pragma unroll
    for (int v = 0; v < 8; ++v)
        Crow[(size_t)(v + 8 * hi) * DM_] = acc[v];
}

// ---------------------------------------------------------------------------
extern "C" void kernel_launch(void* const* d_in, const int* in_sizes, int n_in,
                              void* d_out, int out_size, void* d_ws, size_t ws_size,
                              hipStream_t stream) {
    const float* query = (const float*)d_in[0];
    const float* key   = (const float*)d_in[1];
    const float* value = (const float*)d_in[2];
    const float* w_q   = (const float*)d_in[3];
    const float* w_k   = (const float*)d_in[4];
    const float* w_v   = (const float*)d_in[5];
    const float* v_a   = (const float*)d_in[6];
    const float* w_o   = (const float*)d_in[7];

    const size_t PROJ = (size_t)B_ * T_ * DM_;   // 524288 floats
    float* out  = (float*)d_out;                 // (B,T,256)
    float* attn = (float*)d_out + PROJ;          // (B,H,T,S)

    float* ws  = (float*)d_ws;
    float* Q   = ws;                             // (B*T,256)
    float* Kp  = ws + PROJ;                      // (B*S,256)
    float* Vp  = ws + 2 * PROJ;                  // (B*S,256)
    float* Att = ws + 3 * PROJ;                  // attended (B*T,256)

    const int M = B_ * T_;                       // 2048
    const int tiles = (M / 16) * (DM_ / 16);     // 2048 tiles
    const int gblocks = tiles / 4;               // 4 waves (tiles) per block

    // Q/K/V projections: x @ W^T (weights stored (out,in))
    gemm_wmma_f32<true><<<gblocks, 128, 0, stream>>>(query, w_q, Q,  M, DM_, DM_, DM_, DM_, DM_);
    gemm_wmma_f32<true><<<gblocks, 128, 0, stream>>>(key,   w_k, Kp, M, DM_, DM_, DM_, DM_, DM_);
    gemm_wmma_f32<true><<<gblocks, 128, 0, stream>>>(value, w_v, Vp, M, DM_, DM_, DM_, DM_, DM_);

    // additive scores + softmax -> attn (second output, written in-place)
    scores_softmax<<<dim3(T_, B_ * H_), 256, 0, stream>>>(Q, Kp, v_a, attn);

    // attended = attn @ V, head-interleaved to (B,T,256)
    attended_wmma<<<dim3(T_ / 16, B_ * H_), 128, 0, stream>>>(attn, Vp, Att);

    // out = attended @ w_o^T
    gemm_wmma_f32<true><<<gblocks, 128, 0, stream>>>(Att, w_o, out, M, DM_, DM_, DM_, DM_, DM_);
}